// DynamicThresholdPooling_50637664420180
// MI455X (gfx1250) — compile-verified
//
#include <hip/hip_runtime.h>
#include <cstdint>
#include <cstddef>

// CDNA5 TDM descriptor-group vector types (probe-confirmed signatures)
typedef unsigned int u32x4 __attribute__((ext_vector_type(4)));
typedef int          i32x4 __attribute__((ext_vector_type(4)));
typedef int          i32x8 __attribute__((ext_vector_type(8)));

#define PCT 0.90
#define NWAVES 32          // 1024 threads / wave32
#define HPAD 257           // 257*4B: same bin of different waves -> different LDS banks

__device__ __forceinline__ unsigned orderKey(float f) {
    unsigned u = __float_as_uint(f);
    return (u & 0x80000000u) ? ~u : (u | 0x80000000u);
}
__device__ __forceinline__ float keyToFloat(unsigned k) {
    unsigned u = (k & 0x80000000u) ? (k & 0x7FFFFFFFu) : ~k;
    return __uint_as_float(u);
}

__global__ __launch_bounds__(1024)
void dtp_kernel(const float* __restrict__ in, float* __restrict__ out, int N) {
    extern __shared__ unsigned smemU[];              // N order-keys (256 KB for N=65536)
    __shared__ int      whist[NWAVES][HPAD];         // per-wave bank-skewed histograms
    __shared__ int      hist[256];                   // combined histogram
    __shared__ int      s_bin, s_before;
    __shared__ int      s_need, s_cnt;
    __shared__ unsigned s_mingt;
    __shared__ float    s_thresh;
    __shared__ double   s_sum;

    const int tid = threadIdx.x;
    const int bs  = blockDim.x;
    const int row = blockIdx.x;
    const float* rowp = in + (size_t)row * (size_t)N;

    // ---- Stage 0: CDNA5 Tensor Data Mover copy global -> LDS (wave 0 issues) ----
    // LDS bytes are fully overwritten by the keyed cooperative pass below (after
    // s_wait_tensorcnt + barrier), so correctness never depends on D# semantics.
    if ((tid >> 5) == 0 && (N & 4095) == 0 && (N >> 12) <= 0xFFFF) {
        unsigned lds_base = (unsigned)(size_t)smemU;               // LDS byte offset
        unsigned long long ga = (unsigned long long)(size_t)rowp;  // global byte addr
        unsigned d0 = 4096u;                 // tile/tensor dim0 (elements)
        unsigned d1 = (unsigned)(N >> 12);   // tile/tensor dim1 (rows)
        // D# group0: count=1 | lds_addr | global_addr[56:0] | type=2 (ISA 8.3)
        u32x4 g0 = { 1u,
                     lds_base,
                     (unsigned)(ga & 0xFFFFFFFFull),
                     (unsigned)((ga >> 32) & 0x01FFFFFFull) | 0x80000000u };
        // D# group1: data_size=4B; tensor_dim0=d0; tensor_dim1=d1; tile_dim0=d0;
        //            tile_dim1=d1; tensor_dim0_stride=d0 (ISA 8.4 bit layout)
        i32x8 g1 = { (int)(2u << 16),
                     (int)((d0 & 0xFFFFu) << 16),
                     (int)((d0 >> 16) | ((d1 & 0xFFFFu) << 16)),
                     (int)((d1 >> 16) | ((d0 & 0xFFFFu) << 16)),
                     (int)(d1 & 0xFFFFu),
                     (int)d0, 0, 0 };
        i32x4 g2 = { 0, 0, 0, 0 };
        i32x4 g3 = { 0, 0, 0, 0 };
#if __clang_major__ >= 23
        i32x8 g4 = { 0, 0, 0, 0, 0, 0, 0, 0 };
        __builtin_amdgcn_tensor_load_to_lds(g0, g1, g2, g3, g4, 0);
#else
        __builtin_amdgcn_tensor_load_to_lds(g0, g1, g2, g3, 0);
#endif
        __builtin_amdgcn_s_wait_tensorcnt(0);
    }
    __syncthreads();

    // ---- Stage 1: authoritative cooperative load, converting to order-keys ----
    if ((N & 3) == 0) {
        const float4* rp4 = (const float4*)rowp;
        uint4*        sp4 = (uint4*)smemU;
        int n4 = N >> 2;
        for (int i = tid; i < n4; i += bs) {
            float4 v = rp4[i];
            uint4 kk;
            kk.x = orderKey(v.x); kk.y = orderKey(v.y);
            kk.z = orderKey(v.z); kk.w = orderKey(v.w);
            sp4[i] = kk;
        }
    } else {
        for (int i = tid; i < N; i += bs) smemU[i] = orderKey(rowp[i]);
    }
    __syncthreads();

    // quantile position: k-th and (k+1)-th order statistics, linear interpolation
    double pos = PCT * (double)(N - 1);
    int   k    = (int)pos;
    float frac = (float)(pos - (double)k);

    // ---- Stage 2: 4-pass radix select (8 bits/pass), LDS-resident ----
    unsigned prefix = 0;
    int rank = k;                       // rank within prefix-matching subset
    for (int shift = 24; shift >= 0; shift -= 8) {
        unsigned highmask = (shift == 24) ? 0u : (0xFFFFFFFFu << (shift + 8));
        for (int i = tid; i < NWAVES * HPAD; i += bs) ((int*)whist)[i] = 0;
        __syncthreads();
        int* myh = whist[tid >> 5];
        for (int i = tid; i < N; i += bs) {
            unsigned key = smemU[i];
            if ((key & highmask) == prefix)
                atomicAdd(&myh[(key >> shift) & 0xFFu], 1);
        }
        __syncthreads();
        if (tid < 256) {
            int s = 0;
            #pragma unroll
            for (int w = 0; w < NWAVES; ++w) s += whist[w][tid];
            hist[tid] = s;
        }
        __syncthreads();
        if (tid == 0) {
            int cum = 0, b = 0;
            for (; b < 255; ++b) {
                int c = hist[b];
                if (cum + c > rank) break;
                cum += c;
            }
            s_bin = b; s_before = cum;
        }
        __syncthreads();
        prefix |= ((unsigned)s_bin) << shift;
        rank   -= s_before;
        __syncthreads();
    }
    const unsigned ukey = prefix;       // key of k-th smallest; rank = k - count(<ukey)

    // ---- Stage 3: successor from the final-pass histogram (no extra sweep) ----
    // hist[] still holds counts of low byte among keys matching ukey's top 24 bits:
    //   c_eq = hist[s_bin]; successor (if in same top-24 group) = next non-empty bin.
    if (tid == 0) {
        int   ceq = hist[s_bin];
        float xk  = keyToFloat(ukey);
        s_need  = 0;
        s_mingt = 0xFFFFFFFFu;
        s_sum   = 0.0;
        s_cnt   = 0;
        if ((k + 1) >= N || (rank + 1) < ceq) {
            s_thresh = xk;                       // x_{k+1} == x_k
        } else {
            int b = s_bin + 1;
            for (; b < 256 && hist[b] == 0; ++b) {}
            if (b < 256) {
                float xk1 = keyToFloat((ukey & 0xFFFFFF00u) | (unsigned)b);
                s_thresh = xk + frac * (xk1 - xk);
            } else {
                s_need = 1;                      // successor lives in another top-24 group
            }
        }
    }
    __syncthreads();
    if (s_need) {                                // rare fallback: global min key > ukey
        unsigned mymin = 0xFFFFFFFFu;
        for (int i = tid; i < N; i += bs) {
            unsigned key = smemU[i];
            if (key > ukey && key < mymin) mymin = key;
        }
        atomicMin(&s_mingt, mymin);
        __syncthreads();
        if (tid == 0) {
            float xk  = keyToFloat(ukey);
            float xk1 = (s_mingt != 0xFFFFFFFFu) ? keyToFloat(s_mingt) : xk;
            s_thresh = xk + frac * (xk1 - xk);
        }
        __syncthreads();
    }

    // ---- Stage 4: masked mean over x > threshold ----
    {
        float t = s_thresh;
        double mysum = 0.0; int mycnt = 0;
        for (int i = tid; i < N; i += bs) {
            float x = keyToFloat(smemU[i]);
            if (x > t) { mysum += (double)x; mycnt++; }
        }
        atomicAdd(&s_sum, mysum);
        atomicAdd(&s_cnt, mycnt);
    }
    __syncthreads();
    if (tid == 0) {
        int c = (s_cnt > 0) ? s_cnt : 1;
        out[row] = (float)(s_sum / (double)c);
    }
}

extern "C" void kernel_launch(void* const* d_in, const int* in_sizes, int n_in,
                              void* d_out, int out_size, void* d_ws, size_t ws_size,
                              hipStream_t stream) {
    (void)n_in; (void)d_ws; (void)ws_size;
    const float* in  = (const float*)d_in[0];
    float*       out = (float*)d_out;
    int B = out_size;                    // 512 rows
    if (B <= 0) return;
    int N = in_sizes[0] / B;             // 65536 cols
    if (N <= 0) return;
    size_t shmem = (size_t)N * sizeof(unsigned);  // row resident in 320KB WGP LDS
    dim3 grid(B), block(1024);
    dtp_kernel<<<grid, block, shmem, stream>>>(in, out, N);
}